// QSSMGatedBlock_32762010534138
// MI455X (gfx1250) — compile-verified
//
#include <hip/hip_runtime.h>
#include <hip/hip_bf16.h>
#include <math.h>

// ---------------- problem constants (from reference) ----------------
#define E_DIM   1024
#define DIN     2048
#define N_ST    16          // SSM state dim N
#define K_CONV  4
#define DTR     64
#define B_SZ    2
#define L_SEQ   1024
#define R_TT    16
#define BL      (B_SZ * L_SEQ)   // 2048 flattened tokens

// ---------------- WMMA vector types ----------------
typedef __attribute__((ext_vector_type(2))) float v2f;
typedef __attribute__((ext_vector_type(8))) float v8f;

// =====================================================================
// 1) Build dense weight from TT cores:
//    W[(i*m2+j), (u*n2+v)] = sum_r g1[i,u,r] * g2[r,j,v]
//    g1: [m1, n1, R]   g2: [R, m2, n2]   W: [m1*m2, n1*n2] row-major
// =====================================================================
__global__ __launch_bounds__(256)
void build_w_kernel(const float* __restrict__ g1, const float* __restrict__ g2,
                    float* __restrict__ W, int m1, int m2, int n1, int n2)
{
    size_t total = (size_t)m1 * m2 * n1 * n2;
    size_t idx = (size_t)blockIdx.x * 256 + threadIdx.x;
    if (idx >= total) return;
    int Nout = n1 * n2;
    size_t rowi = idx / Nout;
    int colo = (int)(idx % Nout);
    int i = (int)(rowi / m2), j = (int)(rowi % m2);
    int u = colo / n2, v = colo % n2;
    const float* g1p = g1 + ((size_t)i * n1 + u) * R_TT;
    float s = 0.f;
#pragma unroll
    for (int r = 0; r < R_TT; ++r)
        s += g1p[r] * g2[((size_t)r * m2 + j) * n2 + v];
    W[idx] = s;
}

// =====================================================================
// 2) fp32 WMMA GEMM:  C[M,N] = act( A[M,K](lda) * Bw[K,N] + bias[N] )
//    Block: 256 threads (8 waves). Block tile: 16(M) x 256(N).
//    Each wave owns a 16x32 tile (two v8f accumulators) so every A
//    fragment feeds two v_wmma_f32_16x16x4_f32.
//    LDS holds K-chunks (KC=64) with A and B stored as K-pair float2 so
//    every fragment operand is a single ds_load_b64.
//    act: 0 = none, 1 = softplus
//    Requires: M%16==0, K%64==0, N%16==0.
// =====================================================================
#define KC 64
#define TN 256
#define KP (KC / 2)          // K pairs per chunk

__device__ __forceinline__ float softplus_f(float v) {
    return (v > 20.f) ? v : log1pf(expf(v));
}

__global__ __launch_bounds__(256)
void gemm_wmma_f32(const float* __restrict__ A, int lda,
                   const float* __restrict__ Bw,
                   const float* __restrict__ bias,
                   float* __restrict__ C,
                   int M, int N, int K, int act)
{
    // sA2[p*17 + r]  = { A[r][2p], A[r][2p+1] }   (pad 17 -> bank-spread b64 reads)
    // sB2[p*TN + c]  = { B[2p][c], B[2p+1][c] }
    __shared__ v2f sA2[KP * 17];
    __shared__ v2f sB2[KP * TN];

    const int tid  = threadIdx.x;
    const int wave = tid >> 5;
    const int lane = tid & 31;
    const int hl   = lane >> 4;         // half-wave: 0 or 1
    const int l16  = lane & 15;

    const int tm  = blockIdx.y * 16;    // M tile origin
    const int tn0 = blockIdx.x * TN;    // block N origin
    const int wn  = tn0 + wave * 32;    // this wave's N origin (2 sub-tiles)
    const int wc  = wave * 32;          // wave's col offset inside LDS tile

    v8f acc0 = {0.f, 0.f, 0.f, 0.f, 0.f, 0.f, 0.f, 0.f};
    v8f acc1 = {0.f, 0.f, 0.f, 0.f, 0.f, 0.f, 0.f, 0.f};

    for (int k0 = 0; k0 < K; k0 += KC) {
        // prefetch next chunk into L2/L1 (global_prefetch_b8)
        if (k0 + KC < K) {
            __builtin_prefetch(&A[(size_t)(tm + (tid & 15)) * lda + k0 + KC], 0, 0);
            int pc = tn0 + tid;
            if (pc < N) __builtin_prefetch(&Bw[(size_t)(k0 + KC) * N + pc], 0, 0);
        }
        // stage A: 16 rows x KC cols as K-pair float2 (2 pairs per thread)
        for (int i = tid; i < 16 * KP; i += 256) {
            int r = i >> 5;             // row 0..15
            int pc = i & 31;            // K-pair 0..31
            const float* ap = A + (size_t)(tm + r) * lda + k0 + 2 * pc;
            v2f av; av[0] = ap[0]; av[1] = ap[1];   // contiguous -> b64 load
            sA2[pc * 17 + r] = av;
        }
        // stage B: KC rows x TN cols as K-pair float2
        for (int i = tid; i < KP * TN; i += 256) {
            int p = i >> 8;             // K-pair 0..31
            int c = i & (TN - 1);       // col 0..255
            int gc = tn0 + c;
            v2f bv;
            if (gc < N) {
                bv[0] = Bw[(size_t)(k0 + 2 * p) * N + gc];
                bv[1] = Bw[(size_t)(k0 + 2 * p + 1) * N + gc];
            } else {
                bv[0] = 0.f; bv[1] = 0.f;
            }
            sB2[p * TN + c] = bv;
        }
        __syncthreads();

        if (wn < N) {
#pragma unroll
            for (int kk = 0; kk < KC; kk += 4) {
                const int prow = (kk >> 1) + hl;       // K-pair for this half-wave
                v2f a  = sA2[prow * 17 + l16];         // one ds_load_b64
                v2f b0 = sB2[prow * TN + wc + l16];    // one ds_load_b64
                v2f b1 = sB2[prow * TN + wc + 16 + l16];
                acc0 = __builtin_amdgcn_wmma_f32_16x16x4_f32(
                        false, a, false, b0, (short)0, acc0, false, false);
                acc1 = __builtin_amdgcn_wmma_f32_16x16x4_f32(
                        false, a, false, b1, (short)0, acc1, false, false);
            }
        }
        __syncthreads();
    }

    // C/D layout: VGPR g -> M = g (lanes 0-15) / g+8 (lanes 16-31), N = lane%16
    const int rbase = tm + (hl ? 8 : 0);
    if (wn < N) {
        const int col = wn + l16;
        const float bv = bias[col];
#pragma unroll
        for (int g = 0; g < 8; ++g) {
            float v = acc0[g] + bv;
            if (act == 1) v = softplus_f(v);
            C[(size_t)(rbase + g) * N + col] = v;
        }
    }
    if (wn + 16 < N) {
        const int col = wn + 16 + l16;
        const float bv = bias[col];
#pragma unroll
        for (int g = 0; g < 8; ++g) {
            float v = acc1[g] + bv;
            if (act == 1) v = softplus_f(v);
            C[(size_t)(rbase + g) * N + col] = v;
        }
    }
}

// =====================================================================
// 3) depthwise conv (K=4, SAME: pad_lo=1) + SiLU * z
//    xz: [B, L, 2*DIN] ; y: [B, L, DIN]
// =====================================================================
__global__ __launch_bounds__(256)
void conv_gate_kernel(const float* __restrict__ xz,
                      const float* __restrict__ cw,   // [K,1,DIN]
                      const float* __restrict__ cb,   // [DIN]
                      float* __restrict__ y)
{
    size_t idx = (size_t)blockIdx.x * 256 + threadIdx.x;
    if (idx >= (size_t)BL * DIN) return;
    int d = (int)(idx % DIN);
    size_t bl = idx / DIN;
    int t = (int)(bl % L_SEQ);
    int b = (int)(bl / L_SEQ);
    size_t rowbase = (size_t)b * L_SEQ;

    float s = 0.f;
#pragma unroll
    for (int k = 0; k < K_CONV; ++k) {
        int tt = t + k - 1;                       // SAME, pad_lo = 1
        if (tt >= 0 && tt < L_SEQ)
            s += xz[(rowbase + tt) * (2 * DIN) + d] * cw[k * DIN + d];
    }
    s += cb[d];
    float sil = s / (1.f + expf(-s));
    float z = xz[(rowbase + t) * (2 * DIN) + DIN + d];
    y[idx] = sil * z;
}

// =====================================================================
// 4) SSM scan: 16 lanes per (b,d) channel; serial over L.
//    h' = exp(dt*A)*h + dt*u*B ;  y = sum_n h*C + D*u
//    partials[block] = sum of all h over steps (deterministic reduction)
// =====================================================================
__global__ __launch_bounds__(256)
void ssm_scan_kernel(const float* __restrict__ u_in,   // y gated [BL, DIN]
                     const float* __restrict__ dtv,    // [BL, DIN]
                     const float* __restrict__ dtBC,   // [BL, 96]
                     const float* __restrict__ A_log,  // [DIN, N]
                     const float* __restrict__ D_skip, // [DIN]
                     float* __restrict__ ssm,          // [BL, DIN]
                     float* __restrict__ partials)
{
    const int tid = threadIdx.x;
    const int idx = blockIdx.x * 256 + tid;    // over B*DIN*N = 65536
    const int n  = idx & 15;
    const int bd = idx >> 4;
    const int d  = bd & (DIN - 1);
    const int b  = bd / DIN;

    const float Adn = -expf(A_log[d * N_ST + n]);
    const float Dd  = D_skip[d];
    const size_t base = (size_t)b * L_SEQ;

    float h = 0.f, hsum = 0.f;
    for (int t = 0; t < L_SEQ; ++t) {
        const size_t row = base + t;
        const float dt = dtv[row * DIN + d];
        const float u  = u_in[row * DIN + d];
        const float Bn = dtBC[row * 96 + DTR + n];
        const float Cn = dtBC[row * 96 + DTR + N_ST + n];
        h = expf(dt * Adn) * h + dt * u * Bn;
        hsum += h;
        float c = h * Cn;
        c += __shfl_xor(c, 1, 16);
        c += __shfl_xor(c, 2, 16);
        c += __shfl_xor(c, 4, 16);
        c += __shfl_xor(c, 8, 16);
        if (n == 0) ssm[row * DIN + d] = c + Dd * u;
    }

    __shared__ float red[256];
    red[tid] = hsum;
    __syncthreads();
    for (int s = 128; s > 0; s >>= 1) {
        if (tid < s) red[tid] += red[tid + s];
        __syncthreads();
    }
    if (tid == 0) partials[blockIdx.x] = red[0];
}

__global__ __launch_bounds__(256)
void reduce_partials_kernel(const float* __restrict__ partials, int n,
                            float* __restrict__ scalar)
{
    __shared__ float red[256];
    const int tid = threadIdx.x;
    float s = 0.f;
    for (int i = tid; i < n; i += 256) s += partials[i];
    red[tid] = s;
    __syncthreads();
    for (int k = 128; k > 0; k >>= 1) {
        if (tid < k) red[tid] += red[tid + k];
        __syncthreads();
    }
    if (tid == 0)
        *scalar = red[0] / ((float)L_SEQ * B_SZ * DIN * N_ST) * 1e-8f;
}

__global__ __launch_bounds__(256)
void add_scalar_kernel(float* __restrict__ ssm, const float* __restrict__ scalar,
                       size_t n)
{
    size_t i = (size_t)blockIdx.x * 256 + threadIdx.x;
    if (i < n) ssm[i] += *scalar;
}

// =====================================================================
// launch
// =====================================================================
extern "C" void kernel_launch(void* const* d_in, const int* in_sizes, int n_in,
                              void* d_out, int out_size, void* d_ws, size_t ws_size,
                              hipStream_t stream)
{
    const float* x       = (const float*)d_in[0];
    const float* in_g1   = (const float*)d_in[1];
    const float* in_g2   = (const float*)d_in[2];
    const float* in_b    = (const float*)d_in[3];
    const float* xp_g1   = (const float*)d_in[4];
    const float* xp_g2   = (const float*)d_in[5];
    const float* xp_b    = (const float*)d_in[6];
    const float* dt_g1   = (const float*)d_in[7];
    const float* dt_g2   = (const float*)d_in[8];
    const float* dt_b    = (const float*)d_in[9];
    const float* out_g1  = (const float*)d_in[10];
    const float* out_g2  = (const float*)d_in[11];
    const float* out_b   = (const float*)d_in[12];
    const float* conv_w  = (const float*)d_in[13];
    const float* conv_b  = (const float*)d_in[14];
    const float* A_log   = (const float*)d_in[15];
    const float* D_skip  = (const float*)d_in[16];

    float* ws = (float*)d_ws;
    size_t o = 0;
    float* W_in  = ws + o; o += (size_t)E_DIM * (2 * DIN);   // 1024 x 4096
    float* W_xp  = ws + o; o += (size_t)DIN * 96;            // 2048 x 96
    float* W_dt  = ws + o; o += (size_t)DTR * DIN;           // 64 x 2048
    float* W_out = ws + o; o += (size_t)DIN * E_DIM;         // 2048 x 1024
    float* xz    = ws + o; o += (size_t)BL * (2 * DIN);      // 2048 x 4096
    float* yb    = ws + o; o += (size_t)BL * DIN;            // 2048 x 2048
    float* dtBC  = ws + o; o += (size_t)BL * 96;             // 2048 x 96
    float* dtb   = ws + o; o += (size_t)BL * DIN;            // 2048 x 2048
    float* ssm   = ws + o; o += (size_t)BL * DIN;            // 2048 x 2048
    float* parts = ws + o; o += 256;
    float* scal  = ws + o; o += 1;
    (void)ws_size; (void)n_in; (void)in_sizes; (void)out_size;

    // ---- 1) materialize dense TT weights ----
    {
        size_t t;
        t = (size_t)E_DIM * (2 * DIN);
        build_w_kernel<<<(t + 255) / 256, 256, 0, stream>>>(in_g1, in_g2, W_in, 32, 32, 64, 64);
        t = (size_t)DIN * 96;
        build_w_kernel<<<(t + 255) / 256, 256, 0, stream>>>(xp_g1, xp_g2, W_xp, 32, 64, 8, 12);
        t = (size_t)DTR * DIN;
        build_w_kernel<<<(t + 255) / 256, 256, 0, stream>>>(dt_g1, dt_g2, W_dt, 8, 8, 32, 64);
        t = (size_t)DIN * E_DIM;
        build_w_kernel<<<(t + 255) / 256, 256, 0, stream>>>(out_g1, out_g2, W_out, 32, 64, 32, 32);
    }

    // ---- 2) in-proj: xz = x * W_in + in_b   [2048,4096] ----
    {
        dim3 grid((2 * DIN + TN - 1) / TN, BL / 16);
        gemm_wmma_f32<<<grid, 256, 0, stream>>>(x, E_DIM, W_in, in_b, xz,
                                                BL, 2 * DIN, E_DIM, 0);
    }

    // ---- 3) conv + silu gate -> yb ----
    {
        size_t t = (size_t)BL * DIN;
        conv_gate_kernel<<<(t + 255) / 256, 256, 0, stream>>>(xz, conv_w, conv_b, yb);
    }

    // ---- 4) xp-proj: dtBC = yb * W_xp + xp_b  [2048,96] ----
    {
        dim3 grid((96 + TN - 1) / TN, BL / 16);
        gemm_wmma_f32<<<grid, 256, 0, stream>>>(yb, DIN, W_xp, xp_b, dtBC,
                                                BL, 96, DIN, 0);
    }

    // ---- 5) dt-proj: dtb = softplus(dtBC[:, :64] * W_dt + dt_b) [2048,2048] ----
    {
        dim3 grid((DIN + TN - 1) / TN, BL / 16);
        gemm_wmma_f32<<<grid, 256, 0, stream>>>(dtBC, 96, W_dt, dt_b, dtb,
                                                BL, DIN, DTR, 1);
    }

    // ---- 6) SSM scan + deterministic h-mean reduction ----
    ssm_scan_kernel<<<(B_SZ * DIN * N_ST) / 256, 256, 0, stream>>>(
        yb, dtb, dtBC, A_log, D_skip, ssm, parts);
    reduce_partials_kernel<<<1, 256, 0, stream>>>(parts, 256, scal);
    {
        size_t t = (size_t)BL * DIN;
        add_scalar_kernel<<<(t + 255) / 256, 256, 0, stream>>>(ssm, scal, t);
    }

    // ---- 7) out-proj: out = ssm * W_out + out_b  [2048,1024] ----
    {
        dim3 grid((E_DIM + TN - 1) / TN, BL / 16);
        gemm_wmma_f32<<<grid, 256, 0, stream>>>(ssm, DIN, W_out, out_b,
                                                (float*)d_out, BL, E_DIM, DIN, 0);
    }
}